// ResPool_120259084567
// MI455X (gfx1250) — compile-verified
//
#include <hip/hip_runtime.h>
#include <float.h>

typedef __attribute__((ext_vector_type(2))) float v2f;
typedef __attribute__((ext_vector_type(8))) float v8f;

#define LNUM 3
#define CCH  128      // C
#define OUTC 128      // OUT
#define TWOC 256      // 2C

// ---------------------------------------------------------------------------
// Kernel 1: fused max-over-L + contiguous segment-max pooling.
// One block per segment; 8 sub-waves stride over rows, each lane owns 4
// channels via float4 loads (16B per lane per instruction to saturate HBM).
// ---------------------------------------------------------------------------
__global__ __launch_bounds__(256) void segpool_kernel(
    const float* __restrict__ feats,   // [L, N, C]
    const int*   __restrict__ offs,    // [B] strictly increasing, offs[0]=0
    float*       __restrict__ pool,    // [B, C] workspace output
    int Nrows, int Bseg)
{
  const int b     = blockIdx.x;
  const int start = offs[b];
  const int end   = (b == Bseg - 1) ? Nrows : offs[b + 1];
  const int lane  = threadIdx.x & 31;
  const int sub   = threadIdx.x >> 5;      // 0..7
  const int c0    = lane << 2;             // 4 channels per lane

  float4 acc = make_float4(-FLT_MAX, -FLT_MAX, -FLT_MAX, -FLT_MAX);
  for (int r = start + sub; r < end; r += 8) {
#pragma unroll
    for (int l = 0; l < LNUM; ++l) {
      const float4 v = *(const float4*)(feats + ((size_t)l * Nrows + r) * CCH + c0);
      acc.x = fmaxf(acc.x, v.x);
      acc.y = fmaxf(acc.y, v.y);
      acc.z = fmaxf(acc.z, v.z);
      acc.w = fmaxf(acc.w, v.w);
    }
  }

  __shared__ float sm[8][CCH];
  sm[sub][c0 + 0] = acc.x;
  sm[sub][c0 + 1] = acc.y;
  sm[sub][c0 + 2] = acc.z;
  sm[sub][c0 + 3] = acc.w;
  __syncthreads();

  if (threadIdx.x < CCH) {
    float m = sm[0][threadIdx.x];
#pragma unroll
    for (int s = 1; s < 8; ++s) m = fmaxf(m, sm[s][threadIdx.x]);
    pool[(size_t)b * CCH + threadIdx.x] = m;
  }
}

// ---------------------------------------------------------------------------
// Kernel 2: gather root rows (max over L), f32 WMMA GEMM [16,256]x[256,128],
// fused bias + ReLU + LayerNorm. One block = 16 output rows, 8 waves, each
// wave owns one 16x16 output tile (64 chained v_wmma_f32_16x16x4_f32).
// ---------------------------------------------------------------------------
__global__ __launch_bounds__(256) void gemm_ln_kernel(
    const float* __restrict__ feats,   // [L, N, C]
    const float* __restrict__ W,       // [2C, OUT] row-major
    const float* __restrict__ bias,    // [OUT]
    const float* __restrict__ gamma,   // [OUT]
    const float* __restrict__ beta,    // [OUT]
    const int*   __restrict__ ego,     // [B]
    const float* __restrict__ pool,    // [B, C]
    float*       __restrict__ out,     // [B, OUT]
    int Nrows)
{
  __shared__ float xs[16][TWOC + 1];   // +1 pad to break LDS bank alignment
  __shared__ float hs[16][OUTC];

  const int tid = threadIdx.x;
  const int rb  = blockIdx.x * 16;     // first output row of this block

  // Stage x = [root | pool] into LDS; root = max over L of feats[:, ego, :]
  for (int i = tid; i < 16 * CCH; i += 256) {
    const int row = i >> 7;
    const int c   = i & (CCH - 1);
    const int n   = ego[rb + row];
    const float* fp = feats + (size_t)n * CCH + c;
    float v = fp[0];
    v = fmaxf(v, fp[(size_t)Nrows * CCH]);
    v = fmaxf(v, fp[(size_t)2 * Nrows * CCH]);
    xs[row][c]       = v;
    xs[row][CCH + c] = pool[(size_t)(rb + row) * CCH + c];
  }
  __syncthreads();

  const int wave = tid >> 5;           // 0..7 -> output column tile
  const int lane = tid & 31;
  const int half = lane >> 4;          // 0: lanes 0-15, 1: lanes 16-31
  const int l16  = lane & 15;
  const int col  = wave * 16 + l16;    // output column of this lane

  // D = A(16x4) x B(4x16) + C, K stepped by 4 up to 256.
  // A layout (ISA 7.12.2, 32-bit A 16x4): lanes 0-15 hold K=0(v0),K=1(v1);
  // lanes 16-31 hold K=2(v0),K=3(v1); M = lane%16.
  // B layout mirrored: lane holds column N=lane%16; v0/v1 carry K=2*half,+1.
  v8f acc = {0.f, 0.f, 0.f, 0.f, 0.f, 0.f, 0.f, 0.f};
  for (int kk = 0; kk < TWOC; kk += 4) {
    const int k0 = kk + 2 * half;
    v2f a;
    a.x = xs[l16][k0];
    a.y = xs[l16][k0 + 1];
    v2f bm;
    bm.x = W[(size_t)k0 * OUTC + col];
    bm.y = W[(size_t)(k0 + 1) * OUTC + col];
    acc = __builtin_amdgcn_wmma_f32_16x16x4_f32(
        /*neg_a=*/false, a, /*neg_b=*/false, bm,
        /*c_mod=*/(short)0, acc, /*reuse_a=*/false, /*reuse_b=*/false);
  }

  // C/D layout: VGPR v -> row v (lanes 0-15) or v+8 (lanes 16-31), N=lane%16.
  const float bb = bias[col];
#pragma unroll
  for (int v = 0; v < 8; ++v) {
    const int row = v + 8 * half;
    hs[row][col] = fmaxf(acc[v] + bb, 0.0f);
  }
  __syncthreads();

  // LayerNorm: 2 rows per wave, wave32 shuffle reduction over 128 cols.
#pragma unroll
  for (int rr = 0; rr < 2; ++rr) {
    const int row = wave * 2 + rr;
    const int cc  = lane << 2;
    const float4 hv = *(const float4*)&hs[row][cc];
    float s  = hv.x + hv.y + hv.z + hv.w;
    float s2 = hv.x * hv.x + hv.y * hv.y + hv.z * hv.z + hv.w * hv.w;
#pragma unroll
    for (int off = 16; off > 0; off >>= 1) {
      s  += __shfl_xor(s,  off, 32);
      s2 += __shfl_xor(s2, off, 32);
    }
    const float mu   = s * (1.0f / OUTC);
    const float var  = s2 * (1.0f / OUTC) - mu * mu;
    const float rstd = rsqrtf(var + 1e-9f);
    const float4 g  = *(const float4*)&gamma[cc];
    const float4 be = *(const float4*)&beta[cc];
    float4 o;
    o.x = (hv.x - mu) * rstd * g.x + be.x;
    o.y = (hv.y - mu) * rstd * g.y + be.y;
    o.z = (hv.z - mu) * rstd * g.z + be.z;
    o.w = (hv.w - mu) * rstd * g.w + be.w;
    *(float4*)&out[(size_t)(rb + row) * OUTC + cc] = o;
  }
}

// ---------------------------------------------------------------------------
extern "C" void kernel_launch(void* const* d_in, const int* in_sizes, int n_in,
                              void* d_out, int out_size, void* d_ws, size_t ws_size,
                              hipStream_t stream) {
  const float* feats = (const float*)d_in[0];   // [L,N,C] f32
  const float* W     = (const float*)d_in[1];   // [2C,OUT] f32
  const float* bvec  = (const float*)d_in[2];   // [OUT]
  const float* gamma = (const float*)d_in[3];   // [OUT]
  const float* beta  = (const float*)d_in[4];   // [OUT]
  const int*   ego   = (const int*)d_in[5];     // [B]
  const int*   offs  = (const int*)d_in[6];     // [B]
  float*       out   = (float*)d_out;           // [B,OUT] f32

  const int B = in_sizes[5];
  const int N = in_sizes[0] / (LNUM * CCH);

  float* pool = (float*)d_ws;                   // B*C*4 = 4 MB scratch

  segpool_kernel<<<B, 256, 0, stream>>>(feats, offs, pool, N, B);
  gemm_ln_kernel<<<B / 16, 256, 0, stream>>>(feats, W, bvec, gamma, beta,
                                             ego, pool, out, N);
}